// GraphAttentionLayer_29394756174252
// MI455X (gfx1250) — compile-verified
//
#include <hip/hip_runtime.h>
#include <math.h>

// ---------------------------------------------------------------------------
// Graph attention layer, MI455X (gfx1250, wave32, WMMA + TDM).
// B=2, N=512, EN=128, EE=64, F=64, H=8.  Edges dominate: 134MB in/out.
// Per (b,i) row: TWO pipelined TDM descriptors stream edges[b,i,:,:]
// ([512,64] f32, 128KB) into LDS in 256-row halves; WMMA computes ef in place
// over half 0 while the TDM engine is still streaming half 1
// (s_wait_tensorcnt 1 -> compute -> s_wait_tensorcnt 0).  Scores / dual
// softmax / aggregation stay in LDS; the ELU'd edge row leaves via
// non-temporal f32x4 stores.  f32 WMMA (16x16x4) keeps exact f32 semantics.
// ---------------------------------------------------------------------------

typedef __attribute__((ext_vector_type(2))) float v2f;
typedef __attribute__((ext_vector_type(4))) float v4f;
typedef __attribute__((ext_vector_type(8))) float v8f;
typedef __attribute__((ext_vector_type(4))) unsigned int u32x4;
typedef __attribute__((ext_vector_type(4))) int i32x4;
typedef __attribute__((ext_vector_type(8))) int i32x8;

#define B_  2
#define N_  512
#define EN_ 128
#define EE_ 64
#define F_  64
#define H_  8
#define INV_SQRT_D 0.35355339059327373f   // 1/sqrt(8), same for node & edge

__device__ __forceinline__ float elu1(float x) {
  return x > 0.f ? x : expm1f(x);
}

// WMMA f32 16x16x4 fragment mapping (ISA 7.12.2, 32-bit operands):
//   A: lane l -> M = l&15 ; VGPR v -> K = 2*(l>>4) + v
//   B: lane l -> N = l&15 ; VGPR v -> K = 2*(l>>4) + v   (mirror of A)
//   C/D: lane l -> N = l&15 ; c[r] -> M = r + 8*(l>>4)

// ---------------- Kernel 1: nf = nodes @ node_W + node_b  -------------------
__global__ __launch_bounds__(256) void nf_gemm_kernel(
    const float* __restrict__ nodes, const float* __restrict__ node_W,
    const float* __restrict__ node_b, float* __restrict__ nf) {
  const int tid  = threadIdx.x;
  const int lane = tid & 31;
  const int wave = tid >> 5;
  const int mt   = blockIdx.x * 8 + wave;   // 64 row-tiles over 1024 rows
  const int n    = lane & 15;
  const int mrow = mt * 16 + n;
  const int khalf = (lane >> 4) * 2;

  v8f c0 = {}, c1 = {}, c2 = {}, c3 = {};
#pragma unroll 4
  for (int kk = 0; kk < EN_; kk += 4) {
    v2f a, b0, b1, b2, b3;
    a.x = nodes[mrow * EN_ + kk + khalf];
    a.y = nodes[mrow * EN_ + kk + khalf + 1];
    const float* w0 = node_W + (kk + khalf) * F_;
    const float* w1 = node_W + (kk + khalf + 1) * F_;
    b0.x = w0[n];      b0.y = w1[n];
    b1.x = w0[n + 16]; b1.y = w1[n + 16];
    b2.x = w0[n + 32]; b2.y = w1[n + 32];
    b3.x = w0[n + 48]; b3.y = w1[n + 48];
    c0 = __builtin_amdgcn_wmma_f32_16x16x4_f32(false, a, false, b0, (short)0, c0, false, false);
    c1 = __builtin_amdgcn_wmma_f32_16x16x4_f32(false, a, false, b1, (short)0, c1, false, false);
    c2 = __builtin_amdgcn_wmma_f32_16x16x4_f32(false, a, false, b2, (short)0, c2, false, false);
    c3 = __builtin_amdgcn_wmma_f32_16x16x4_f32(false, a, false, b3, (short)0, c3, false, false);
  }
  const float bi0 = node_b[n], bi1 = node_b[n + 16];
  const float bi2 = node_b[n + 32], bi3 = node_b[n + 48];
#pragma unroll
  for (int r = 0; r < 8; ++r) {
    const int m = mt * 16 + r + 8 * (lane >> 4);
    nf[m * F_ + n]      = c0[r] + bi0;
    nf[m * F_ + n + 16] = c1[r] + bi1;
    nf[m * F_ + n + 32] = c2[r] + bi2;
    nf[m * F_ + n + 48] = c3[r] + bi3;
  }
}

// ---------------- Kernel 2: q = nf @ att_W  (tiny) --------------------------
__global__ void q_kernel(const float* __restrict__ nf,
                         const float* __restrict__ att_W,
                         float* __restrict__ q) {
  const int idx = blockIdx.x * blockDim.x + threadIdx.x;   // 8192
  if (idx >= B_ * N_ * H_) return;
  const int r = idx / H_, h = idx % H_;
  float s = 0.f;
#pragma unroll 8
  for (int k = 0; k < F_; ++k) s += nf[r * F_ + k] * att_W[k * H_ + h];
  q[idx] = s;
}

// ---------------- block softmax over 512 LDS entries ------------------------
__device__ void softmax512(float* s, float* red, int tid) {
  float lm = -3.4e38f;
  for (int j = tid; j < N_; j += 256) lm = fmaxf(lm, s[j]);
  red[tid] = lm; __syncthreads();
  for (int st = 128; st > 0; st >>= 1) {
    if (tid < st) red[tid] = fmaxf(red[tid], red[tid + st]);
    __syncthreads();
  }
  const float mx = red[0]; __syncthreads();
  float ls = 0.f;
  for (int j = tid; j < N_; j += 256) {
    const float e = __expf(s[j] - mx);
    s[j] = e; ls += e;
  }
  red[tid] = ls; __syncthreads();
  for (int st = 128; st > 0; st >>= 1) {
    if (tid < st) red[tid] += red[tid + st];
    __syncthreads();
  }
  const float inv = 1.f / red[0]; __syncthreads();
  for (int j = tid; j < N_; j += 256) s[j] *= inv;
  __syncthreads();
}

#if __has_builtin(__builtin_amdgcn_tensor_load_to_lds)
#define HAVE_TDM 1
// Issue a TDM load of a [rows x 64] f32 tile (row-major, stride 64) into LDS.
__device__ __forceinline__ void tdm_load_rows(const float* gsrc, float* ldst,
                                              int rows) {
  const unsigned long long ga = (unsigned long long)(size_t)gsrc;
  u32x4 g0;
  g0.x = 1u;                                      // count=1, user mode
  g0.y = (unsigned)(size_t)ldst;                  // lds_addr (bytes)
  g0.z = (unsigned)(ga & 0xffffffffull);          // global_addr[31:0]
  g0.w = (unsigned)((ga >> 32) & 0x1ffffffull)    // global_addr[56:32]
         | 0x80000000u;                           // type=2 ("image")
  i32x8 g1;
  g1[0] = (2 << 16);                      // data_size = 4B
  g1[1] = (EE_ << 16);                    // tensor_dim0 = 64
  g1[2] = (int)((unsigned)rows << 16);    // tensor_dim1 = rows
  g1[3] = (EE_ << 16);                    // tile_dim0 = 64
  g1[4] = rows;                           // tile_dim1 = rows
  g1[5] = EE_;                            // tensor_dim0_stride = 64
  g1[6] = 0;
  g1[7] = 0;
  i32x4 gz = {0, 0, 0, 0};
#if __clang_major__ >= 23
  i32x8 gz8 = {0, 0, 0, 0, 0, 0, 0, 0};
  __builtin_amdgcn_tensor_load_to_lds(g0, g1, gz, gz, gz8, 0);
#else
  __builtin_amdgcn_tensor_load_to_lds(g0, g1, gz, gz, 0);
#endif
}
// SIMM16 immediate must be a compile-time constant -> template parameter.
template <int CNT>
__device__ __forceinline__ void tdm_wait() {
#if __has_builtin(__builtin_amdgcn_s_wait_tensorcnt)
  __builtin_amdgcn_s_wait_tensorcnt(CNT);
#else
  if (CNT == 0) asm volatile("s_wait_tensorcnt 0x0" ::: "memory");
  else          asm volatile("s_wait_tensorcnt 0x1" ::: "memory");
#endif
}
#else
#define HAVE_TDM 0
#endif

// One 16-row j-tile of the in-place ef GEMM: ef[16x64] = staged @ ews + bias.
__device__ __forceinline__ void ef_tile_gemm(float* ef, const float* ews,
                                             const float* __restrict__ edge_b,
                                             int jt, int lane) {
  const int khalf = (lane >> 4) * 2;
  const int n = lane & 15;
  const int jrow = jt * 16 + n;
  v8f c0 = {}, c1 = {}, c2 = {}, c3 = {};
#pragma unroll 4
  for (int kk = 0; kk < EE_; kk += 4) {
    v2f a, b0, b1, b2, b3;
    a.x = ef[jrow * F_ + kk + khalf];
    a.y = ef[jrow * F_ + kk + khalf + 1];
    const float* w0 = ews + (kk + khalf) * F_;
    const float* w1 = ews + (kk + khalf + 1) * F_;
    b0.x = w0[n];      b0.y = w1[n];
    b1.x = w0[n + 16]; b1.y = w1[n + 16];
    b2.x = w0[n + 32]; b2.y = w1[n + 32];
    b3.x = w0[n + 48]; b3.y = w1[n + 48];
    c0 = __builtin_amdgcn_wmma_f32_16x16x4_f32(false, a, false, b0, (short)0, c0, false, false);
    c1 = __builtin_amdgcn_wmma_f32_16x16x4_f32(false, a, false, b1, (short)0, c1, false, false);
    c2 = __builtin_amdgcn_wmma_f32_16x16x4_f32(false, a, false, b2, (short)0, c2, false, false);
    c3 = __builtin_amdgcn_wmma_f32_16x16x4_f32(false, a, false, b3, (short)0, c3, false, false);
  }
  const float bi0 = edge_b[n], bi1 = edge_b[n + 16];
  const float bi2 = edge_b[n + 32], bi3 = edge_b[n + 48];
#pragma unroll
  for (int r = 0; r < 8; ++r) {
    const int j = jt * 16 + r + 8 * (lane >> 4);
    ef[j * F_ + n]      = c0[r] + bi0;   // overwrite own staged block
    ef[j * F_ + n + 16] = c1[r] + bi1;
    ef[j * F_ + n + 32] = c2[r] + bi2;
    ef[j * F_ + n + 48] = c3[r] + bi3;
  }
}

// ---------------- Kernel 3: per-(b,i) fused edge row ------------------------
__global__ __launch_bounds__(256) void edge_row_kernel(
    const float* __restrict__ edges, const float* __restrict__ edge_W,
    const float* __restrict__ edge_b, const float* __restrict__ att_W,
    const unsigned char* __restrict__ node_mask,
    const float* __restrict__ nf, const float* __restrict__ q,
    float* __restrict__ wn_out, float* __restrict__ aef_out,
    float* __restrict__ out_edges) {
  extern __shared__ float smem[];
  float* ef    = smem;              // 512*64: TDM staging, then ef in place
  float* qs    = ef   + N_ * F_;    // 512*8
  float* ews   = qs   + N_ * H_;    // 64*64
  float* se_s  = ews  + EE_ * F_;   // 512
  float* sn_s  = se_s + N_;         // 512
  float* vv    = sn_s + N_;         // 64
  float* nfrow = vv   + F_;         // 64
  float* red   = nfrow + F_;        // 256

  const int tid  = threadIdx.x;
  const int wave = tid >> 5;
  const int lane = tid & 31;
  const int b = blockIdx.x / N_;
  const int i = blockIdx.x % N_;
  const float mask = (float)node_mask[b * N_ + i];
  const float* erow = edges + (size_t)(b * N_ + i) * N_ * EE_;

  // ---- stream edges[b,i,:,:] into LDS as two pipelined 256-row halves ----
#if HAVE_TDM
  if (wave == 0) {
    tdm_load_rows(erow,                ef,                N_ / 2);
    tdm_load_rows(erow + (N_/2) * EE_, ef + (N_/2) * F_,  N_ / 2);
  }
#else
  for (int t = tid * 4; t < N_ * EE_; t += 1024)
    *(v4f*)(ef + t) = __builtin_nontemporal_load((const v4f*)(erow + t));
#endif

  // stage small operands (overlaps the tensor DMA)
  for (int t = tid; t < EE_ * F_; t += 256) ews[t] = edge_W[t];
  for (int t = tid * 4; t < N_ * H_; t += 1024)
    *(v4f*)(qs + t) = *(const v4f*)(q + b * N_ * H_ + t);
  if (tid < F_) nfrow[tid] = nf[(b * N_ + i) * F_ + tid];

#if HAVE_TDM
  if (wave == 0) tdm_wait<1>();  // half 0 resident; half 1 still in flight
#endif
  __syncthreads();

  // v = att_W @ q_i (waves 0..1: wave-uniform branch)
  if (tid < F_) {
    float s = 0.f;
#pragma unroll
    for (int h = 0; h < H_; ++h) s += att_W[tid * H_ + h] * qs[i * H_ + h];
    vv[tid] = s;
  }

  // GEMM half 0 (j-tiles 0..15) while TDM streams half 1.
  ef_tile_gemm(ef, ews, edge_b, wave,     lane);
  ef_tile_gemm(ef, ews, edge_b, wave + 8, lane);

#if HAVE_TDM
  if (wave == 0) tdm_wait<0>();  // half 1 resident
#endif
  __syncthreads();

  // GEMM half 1 (j-tiles 16..31).
  ef_tile_gemm(ef, ews, edge_b, wave + 16, lane);
  ef_tile_gemm(ef, ews, edge_b, wave + 24, lane);
  __syncthreads();

  // scores: se[j] = (ef[j,:] . v)/sqrt(8);  sn[j] = (q_j . q_i)/sqrt(8)
  for (int j = tid; j < N_; j += 256) {
    float s_e = 0.f, s_n = 0.f;
#pragma unroll 8
    for (int k = 0; k < F_; ++k) s_e += ef[j * F_ + k] * vv[k];
#pragma unroll
    for (int h = 0; h < H_; ++h) s_n += qs[j * H_ + h] * qs[i * H_ + h];
    se_s[j] = s_e * INV_SQRT_D * mask;
    sn_s[j] = s_n * INV_SQRT_D * mask;
  }
  __syncthreads();

  softmax512(se_s, red, tid);   // se_s -> we
  softmax512(sn_s, red, tid);   // sn_s -> wn

  // wn row to workspace (needed by awn = wn^T @ nf)
  for (int j = tid; j < N_; j += 256)
    wn_out[((size_t)b * N_ + i) * N_ + j] = sn_s[j];

  // aef[k] = sum_j we[j] * ef[j,k]
  if (tid < F_) {
    float s = 0.f;
    for (int j = 0; j < N_; ++j) s += se_s[j] * ef[j * F_ + tid];
    aef_out[(b * N_ + i) * F_ + tid] = s;
  }

  // out_edges row: elu((1+we[j])*ef[j,k] + wn[j]*nf[b,i,k]) as streaming f32x4
  float* orow = out_edges + (size_t)(b * N_ + i) * N_ * EE_;
  for (int t = tid * 4; t < N_ * F_; t += 1024) {
    const int j = t >> 6, k = t & 63;
    const float a1 = 1.f + se_s[j];
    const float a2 = sn_s[j];
    v4f e   = *(v4f*)(ef + t);
    v4f nfv = *(v4f*)(nfrow + k);
    v4f o;
    o.x = elu1(a1 * e.x + a2 * nfv.x);
    o.y = elu1(a1 * e.y + a2 * nfv.y);
    o.z = elu1(a1 * e.z + a2 * nfv.z);
    o.w = elu1(a1 * e.w + a2 * nfv.w);
    __builtin_nontemporal_store(o, (v4f*)(orow + t));
  }
}

// ---------------- Kernel 4: awn[b] = wn[b]^T @ nf[b]  (WMMA, K=512) ---------
__global__ __launch_bounds__(256) void awn_gemm_kernel(
    const float* __restrict__ wn, const float* __restrict__ nf,
    float* __restrict__ awn) {
  const int tid  = threadIdx.x;
  const int lane = tid & 31;
  const int wave = tid >> 5;
  const int tile = blockIdx.x * 8 + wave;   // 64 stripes: b(2) x jt(32)
  const int b  = tile >> 5;
  const int jt = tile & 31;
  const int khalf = (lane >> 4) * 2;
  const int n = lane & 15;
  const int jrow = jt * 16 + n;
  const float* wnb = wn + (size_t)b * N_ * N_;
  const float* nfb = nf + b * N_ * F_;

  v8f c0 = {}, c1 = {}, c2 = {}, c3 = {};
#pragma unroll 4
  for (int kk = 0; kk < N_; kk += 4) {      // K runs over i; A = wn^T
    v2f a, b0, b1, b2, b3;
    a.x = wnb[(kk + khalf)     * N_ + jrow];
    a.y = wnb[(kk + khalf + 1) * N_ + jrow];
    const float* f0 = nfb + (kk + khalf) * F_;
    const float* f1 = nfb + (kk + khalf + 1) * F_;
    b0.x = f0[n];      b0.y = f1[n];
    b1.x = f0[n + 16]; b1.y = f1[n + 16];
    b2.x = f0[n + 32]; b2.y = f1[n + 32];
    b3.x = f0[n + 48]; b3.y = f1[n + 48];
    c0 = __builtin_amdgcn_wmma_f32_16x16x4_f32(false, a, false, b0, (short)0, c0, false, false);
    c1 = __builtin_amdgcn_wmma_f32_16x16x4_f32(false, a, false, b1, (short)0, c1, false, false);
    c2 = __builtin_amdgcn_wmma_f32_16x16x4_f32(false, a, false, b2, (short)0, c2, false, false);
    c3 = __builtin_amdgcn_wmma_f32_16x16x4_f32(false, a, false, b3, (short)0, c3, false, false);
  }
#pragma unroll
  for (int r = 0; r < 8; ++r) {
    const int j = jt * 16 + r + 8 * (lane >> 4);
    awn[(b * N_ + j) * F_ + n]      = c0[r];
    awn[(b * N_ + j) * F_ + n + 16] = c1[r];
    awn[(b * N_ + j) * F_ + n + 32] = c2[r];
    awn[(b * N_ + j) * F_ + n + 48] = c3[r];
  }
}

// ---------------- Kernel 5: out_nodes = elu(nf + awn + aef) -----------------
__global__ void nodes_out_kernel(const float* __restrict__ nf,
                                 const float* __restrict__ awn,
                                 const float* __restrict__ aef,
                                 float* __restrict__ out) {
  const int idx = blockIdx.x * blockDim.x + threadIdx.x;
  if (idx >= B_ * N_ * F_) return;
  out[idx] = elu1(nf[idx] + awn[idx] + aef[idx]);
}

// ---------------------------------------------------------------------------
extern "C" void kernel_launch(void* const* d_in, const int* in_sizes, int n_in,
                              void* d_out, int out_size, void* d_ws, size_t ws_size,
                              hipStream_t stream) {
  (void)in_sizes; (void)n_in; (void)out_size; (void)ws_size;
  const float* nodes  = (const float*)d_in[0];
  const float* edges  = (const float*)d_in[1];
  const unsigned char* node_mask = (const unsigned char*)d_in[2];
  const float* node_W = (const float*)d_in[3];
  const float* node_b = (const float*)d_in[4];
  const float* edge_W = (const float*)d_in[5];
  const float* edge_b = (const float*)d_in[6];
  const float* att_W  = (const float*)d_in[7];

  // workspace: nf | q | wn | aef | awn  (~2.9 MB total)
  float* ws     = (float*)d_ws;
  float* nf_ws  = ws;
  float* q_ws   = nf_ws  + B_ * N_ * F_;
  float* wn_ws  = q_ws   + B_ * N_ * H_;
  float* aef_ws = wn_ws  + (size_t)B_ * N_ * N_;
  float* awn_ws = aef_ws + B_ * N_ * F_;

  float* out_nodes = (float*)d_out;                 // first 2*512*64
  float* out_edges = out_nodes + B_ * N_ * F_;      // then 2*512*512*64

  nf_gemm_kernel<<<8, 256, 0, stream>>>(nodes, node_W, node_b, nf_ws);
  q_kernel<<<(B_ * N_ * H_ + 255) / 256, 256, 0, stream>>>(nf_ws, att_W, q_ws);

  const size_t smem_bytes =
      (size_t)(N_ * F_ + N_ * H_ + EE_ * F_ + N_ + N_ + F_ + F_ + 256) * sizeof(float);
  edge_row_kernel<<<B_ * N_, 256, smem_bytes, stream>>>(
      edges, edge_W, edge_b, att_W, node_mask, nf_ws, q_ws,
      wn_ws, aef_ws, out_edges);

  awn_gemm_kernel<<<8, 256, 0, stream>>>(wn_ws, nf_ws, awn_ws);
  nodes_out_kernel<<<(B_ * N_ * F_ + 255) / 256, 256, 0, stream>>>(
      nf_ws, awn_ws, aef_ws, out_nodes);
}